// DetectionLoss_45071386804524
// MI455X (gfx1250) — compile-verified
//
#include <hip/hip_runtime.h>
#include <math.h>

#define N_PRED  25000
#define N_TGT   64
#define N_BATCH 16
#define TILE    256
#define NBLK    32           // argmax blocks per batch
#define PI_F    3.14159265358979323846f

typedef __attribute__((ext_vector_type(2))) float v2f;
typedef __attribute__((ext_vector_type(8))) float v8f;

// Detect the gfx1250 async global->LDS builtins; fall back to sync path if absent.
#if defined(__has_builtin)
#  if __has_builtin(__builtin_amdgcn_global_load_async_to_lds_b32) && \
      __has_builtin(__builtin_amdgcn_s_wait_asynccnt)
#    define USE_ASYNC_LDS 1
#  endif
#endif
#ifndef USE_ASYNC_LDS
#  define USE_ASYNC_LDS 0
#endif

// focal loss term for a single logit x against target t (0 or 1), matching reference
__device__ __forceinline__ float focal_term(float x, float t) {
    float prob = 1.0f / (1.0f + expf(-x));
    float ce   = fmaxf(x, 0.0f) - x * t + log1pf(expf(-fabsf(x)));
    float p_t  = prob * t + (1.0f - prob) * (1.0f - t);
    p_t        = fminf(fmaxf(p_t, 1e-6f), 1.0f - 1e-6f);
    float a_t  = 0.25f * t + 0.75f * (1.0f - t);
    float om   = 1.0f - p_t;
    return a_t * om * om * ce;
}

#if USE_ASYNC_LDS
// Issue 5 coalesced b32 async loads per thread: tile t's 1280 dwords -> LDS raw buf.
// Addresses clamped into this batch's pred slice; garbage in the tail is masked later.
__device__ __forceinline__ void issue_tile_async(const float* pbase, float* rawbuf, int t) {
    const int tid = threadIdx.x;
    #pragma unroll
    for (int k = 0; k < 5; ++k) {
        int didx = k * TILE + tid;
        int g = t * (TILE * 5) + didx;
        g = min(g, N_PRED * 5 - 1);
        __builtin_amdgcn_global_load_async_to_lds_b32(
            (__attribute__((address_space(1))) int*)(pbase + g),
            (__attribute__((address_space(3))) int*)(rawbuf + didx),
            0, 0);
    }
}
#endif

// ---------------------------------------------------------------------------
// K1: per-(batch,target) argmax of pairwise IoU over preds, tiled through LDS
// with a double-buffered ASYNCcnt pipeline (global_load_async_to_lds_b32).
// grid = (NBLK, 16), block = 256.  thread: target m = tid&63, quarter q = tid>>6
// ---------------------------------------------------------------------------
__global__ __launch_bounds__(256) void k_iou_argmax(
    const float* __restrict__ preds,    // (B, N, 5) cxcywh + conf
    const float* __restrict__ targets,  // (B, M, 4) xyxy
    float* __restrict__ partVal,        // (B, NBLK, 64)
    int*   __restrict__ partIdx)        // (B, NBLK, 64)
{
    const int b   = blockIdx.y;
    const int tid = threadIdx.x;
    const int m   = tid & 63;
    const int q   = tid >> 6;

    __shared__ float4 xyb[TILE];       // converted pred boxes (xyxy)
    __shared__ float  sval[256];
    __shared__ int    sidx[256];
#if USE_ASYNC_LDS
    __shared__ float  rawbuf[2][TILE * 5];   // double-buffered raw staging
#endif

    const float* tb = targets + (size_t)(b * N_TGT + m) * 4;
    const float tx1 = tb[0], ty1 = tb[1], tx2 = tb[2], ty2 = tb[3];
    const float areaT = (tx2 - tx1) * (ty2 - ty1);

    const float* pbase = preds + (size_t)b * N_PRED * 5;
    const int ntiles = (N_PRED + TILE - 1) / TILE;

    float bestV = -1.0f;
    int   bestI = 0;

#if USE_ASYNC_LDS
    int parity = 0;
    issue_tile_async(pbase, rawbuf[0], blockIdx.x);   // prime the pipeline
    for (int t = blockIdx.x; t < ntiles; t += gridDim.x) {
        // raw tile t resident in LDS (all waves' async loads done + visible)
        __builtin_amdgcn_s_wait_asynccnt(0);
        __syncthreads();

        // overlap: kick off tile t+stride into the other buffer now
        const int tn = t + gridDim.x;
        if (tn < ntiles) issue_tile_async(pbase, rawbuf[parity ^ 1], tn);

        // convert raw cxcywh -> xyxy (out-of-range preds get a far-away sliver)
        const int n = t * TILE + tid;
        float cx, cy, w, h;
        if (n < N_PRED) {
            cx = rawbuf[parity][tid * 5 + 0];
            cy = rawbuf[parity][tid * 5 + 1];
            w  = fmaxf(rawbuf[parity][tid * 5 + 2], 1e-4f);
            h  = fmaxf(rawbuf[parity][tid * 5 + 3], 1e-4f);
        } else {
            cx = -1.0e8f; cy = -1.0e8f; w = 1e-4f; h = 1e-4f;
        }
        xyb[tid] = make_float4(cx - 0.5f * w, cy - 0.5f * h,
                               cx + 0.5f * w, cy + 0.5f * h);
        __syncthreads();

        const int n0 = t * TILE + q * 64;
        #pragma unroll 4
        for (int j = 0; j < 64; ++j) {
            const float4 pb = xyb[q * 64 + j];     // broadcast within wave
            float iw = fmaxf(fminf(pb.z, tx2) - fmaxf(pb.x, tx1), 0.0f);
            float ih = fmaxf(fminf(pb.w, ty2) - fmaxf(pb.y, ty1), 0.0f);
            float inter = iw * ih;
            float areaP = (pb.z - pb.x) * (pb.w - pb.y);
            float iou = inter / (areaP + areaT - inter);
            if (iou > bestV) { bestV = iou; bestI = n0 + j; }
        }
        __syncthreads();     // xyb fully consumed before next convert overwrites
        parity ^= 1;
    }
#else
    for (int t = blockIdx.x; t < ntiles; t += gridDim.x) {
        const int n = t * TILE + tid;
        __syncthreads();   // previous tile fully consumed before overwrite
        float cx, cy, w, h;
        if (n < N_PRED) {
            const float* p = pbase + (size_t)n * 5;
            cx = p[0]; cy = p[1];
            w = fmaxf(p[2], 1e-4f); h = fmaxf(p[3], 1e-4f);
        } else {
            cx = -1.0e8f; cy = -1.0e8f; w = 1e-4f; h = 1e-4f;
        }
        xyb[tid] = make_float4(cx - 0.5f * w, cy - 0.5f * h,
                               cx + 0.5f * w, cy + 0.5f * h);
        if (t + (int)gridDim.x < ntiles) {
            __builtin_prefetch(pbase + (size_t)(n + gridDim.x * TILE) * 5, 0, 0);
        }
        __syncthreads();

        const int n0 = t * TILE + q * 64;
        #pragma unroll 4
        for (int j = 0; j < 64; ++j) {
            const float4 pb = xyb[q * 64 + j];
            float iw = fmaxf(fminf(pb.z, tx2) - fmaxf(pb.x, tx1), 0.0f);
            float ih = fmaxf(fminf(pb.w, ty2) - fmaxf(pb.y, ty1), 0.0f);
            float inter = iw * ih;
            float areaP = (pb.z - pb.x) * (pb.w - pb.y);
            float iou = inter / (areaP + areaT - inter);
            if (iou > bestV) { bestV = iou; bestI = n0 + j; }
        }
    }
#endif

    sval[tid] = bestV;
    sidx[tid] = bestI;
    __syncthreads();
    if (tid < 64) {
        float v = sval[tid]; int i = sidx[tid];
        #pragma unroll
        for (int k = 1; k < 4; ++k) {
            float v2 = sval[tid + 64 * k]; int i2 = sidx[tid + 64 * k];
            if (v2 > v || (v2 == v && i2 < i)) { v = v2; i = i2; }
        }
        const int o = (b * gridDim.x + blockIdx.x) * 64 + tid;
        partVal[o] = v;
        partIdx[o] = i;
    }
}

// ---------------------------------------------------------------------------
// K2: final argmax reduce + sequential greedy match + CIoU + focal corrections.
// grid = 16, block = 64 (thread = target m)
// ---------------------------------------------------------------------------
__global__ __launch_bounds__(64) void k_match_box(
    const float* __restrict__ preds,
    const float* __restrict__ targets,
    const float* __restrict__ partVal,
    const int*   __restrict__ partIdx,
    float* __restrict__ boxTerm,
    float* __restrict__ focalCorr,
    int nblk)
{
    const int b = blockIdx.x;
    const int m = threadIdx.x;
    __shared__ float sv[64];
    __shared__ int   si[64];

    float bv = -1.0f; int bi = 0;
    for (int k = 0; k < nblk; ++k) {
        const int o = (b * nblk + k) * 64 + m;
        float v = partVal[o]; int i = partIdx[o];
        if (v > bv || (v == bv && i < bi)) { bv = v; bi = i; }
    }
    sv[m] = bv; si[m] = bi;
    __syncthreads();

    if (m == 0) {
        int matched[64]; int nm = 0;
        float boxSum = 0.0f, corr = 0.0f;
        for (int mm = 0; mm < 64; ++mm) {
            const int   id = si[mm];
            const float v  = sv[mm];
            bool ok = v > 0.2f;                 // IOU_THRESHOLD
            for (int p = 0; p < nm && ok; ++p)
                if (matched[p] == id) ok = false;   // conf_t[best] already 1
            if (!ok) continue;
            matched[nm++] = id;

            // gather pred box (cxcywh -> xyxy with w/h clamp)
            const float* p = preds + ((size_t)b * N_PRED + id) * 5;
            float w = fmaxf(p[2], 1e-4f), h = fmaxf(p[3], 1e-4f);
            float px1 = p[0] - 0.5f * w, py1 = p[1] - 0.5f * h;
            float px2 = p[0] + 0.5f * w, py2 = p[1] + 0.5f * h;
            const float* tb = targets + (size_t)(b * N_TGT + mm) * 4;
            float tx1 = tb[0], ty1 = tb[1], tx2 = tb[2], ty2 = tb[3];

            // CIoU (reference semantics, EPS = 1e-7)
            float iw = fmaxf(fminf(px2, tx2) - fmaxf(px1, tx1), 0.0f);
            float ih = fmaxf(fminf(py2, ty2) - fmaxf(py1, ty1), 0.0f);
            float inter = iw * ih;
            float un = (px2 - px1) * (py2 - py1) + (tx2 - tx1) * (ty2 - ty1) - inter;
            float iou = inter / (un + 1e-7f);
            float cx1 = fminf(px1, tx1), cy1 = fminf(py1, ty1);
            float cx2 = fmaxf(px2, tx2), cy2 = fmaxf(py2, ty2);
            float diag = (cx2 - cx1) * (cx2 - cx1) + (cy2 - cy1) * (cy2 - cy1) + 1e-7f;
            float dcx = px1 + px2 - tx1 - tx2, dcy = py1 + py2 - ty1 - ty2;
            float centers = (dcx * dcx + dcy * dcy) * 0.25f;
            float diou = 1.0f - iou + centers / diag;
            float wp = px2 - px1, hp = py2 - py1, wt = tx2 - tx1, ht = ty2 - ty1;
            float dv = atanf(wt / ht) - atanf(wp / hp);
            float vv = (4.0f / (PI_F * PI_F)) * dv * dv;
            float al = vv / (1.0f - iou + vv + 1e-7f);
            boxSum += diou + al * vv;

            // focal correction: target flips 0 -> 1 at this pred
            float x = p[4];
            corr += focal_term(x, 1.0f) - focal_term(x, 0.0f);
        }
        boxTerm[b]   = (nm > 0) ? boxSum / fmaxf((float)nm, 1.0f) : 0.0f;
        focalCorr[b] = corr;
    }
}

// ---------------------------------------------------------------------------
// K3: sum of focal(logit, target=0) over all N preds per batch.
// v_wmma_f32_16x16x4_f32 with B = ones acts as a 64-way cross-lane accumulator:
// D[m][n] = sum_k A[m][k]*1 + C[m][n]. EXEC stays all-ones (uniform loop bounds,
// clamped-index + multiplicative-mask tail).
// grid = 16, block = 1024 (32 waves)
// ---------------------------------------------------------------------------
__global__ __launch_bounds__(1024) void k_focal_base(
    const float* __restrict__ preds,
    float* __restrict__ confBase)
{
    const int b   = blockIdx.x;
    const int tid = threadIdx.x;
    const float* base = preds + (size_t)b * N_PRED * 5;

    v8f c = {0.f, 0.f, 0.f, 0.f, 0.f, 0.f, 0.f, 0.f};
    v2f ones = {1.0f, 1.0f};

    const int elemsPerIter = 2 * 1024;
    const int iters = (N_PRED + elemsPerIter - 1) / elemsPerIter;
    for (int it = 0; it < iters; ++it) {
        int i0 = it * elemsPerIter + tid * 2;
        int i1 = i0 + 1;
        float m0 = (i0 < N_PRED) ? 1.0f : 0.0f;
        float m1 = (i1 < N_PRED) ? 1.0f : 0.0f;
        int c0 = min(i0, N_PRED - 1);
        int c1 = min(i1, N_PRED - 1);
        float v0 = focal_term(base[(size_t)c0 * 5 + 4], 0.0f) * m0;
        float v1 = focal_term(base[(size_t)c1 * 5 + 4], 0.0f) * m1;
        v2f a = {v0, v1};
        c = __builtin_amdgcn_wmma_f32_16x16x4_f32(
                /*neg_a=*/false, a, /*neg_b=*/false, ones,
                /*c_mod=*/(short)0, c, /*reuse_a=*/false, /*reuse_b=*/false);
    }

    // collapse accumulator tile: per lane sum of its 8 m-rows, then add the
    // opposite half-wave (lanes 0-15 hold m=0..7, lanes 16-31 hold m=8..15)
    float s = c[0] + c[1] + c[2] + c[3] + c[4] + c[5] + c[6] + c[7];
    s += __shfl_xor(s, 16, 32);

    __shared__ float ssum[32];
    if ((tid & 31) == 0) ssum[tid >> 5] = s;
    __syncthreads();
    if (tid == 0) {
        float t = 0.0f;
        for (int w = 0; w < 32; ++w) t += ssum[w];
        confBase[b] = t;
    }
}

// ---------------------------------------------------------------------------
// K4: final combine -> scalar mean over batch
// ---------------------------------------------------------------------------
__global__ void k_finalize(const float* __restrict__ boxTerm,
                           const float* __restrict__ focalCorr,
                           const float* __restrict__ confBase,
                           float* __restrict__ out)
{
    if (threadIdx.x == 0 && blockIdx.x == 0) {
        float acc = 0.0f;
        for (int b = 0; b < N_BATCH; ++b) {
            float conf_loss = (confBase[b] + focalCorr[b]) / (float)N_PRED;
            acc += conf_loss + boxTerm[b];
        }
        out[0] = acc / (float)N_BATCH;
    }
}

extern "C" void kernel_launch(void* const* d_in, const int* in_sizes, int n_in,
                              void* d_out, int out_size, void* d_ws, size_t ws_size,
                              hipStream_t stream) {
    (void)in_sizes; (void)n_in; (void)out_size; (void)ws_size;
    const float* preds   = (const float*)d_in[0];   // (16, 25000, 5) f32
    const float* targets = (const float*)d_in[1];   // (16, 64, 4)    f32
    float* out = (float*)d_out;

    // workspace layout (words): fully rewritten every launch
    float* wsf      = (float*)d_ws;
    float* partVal  = wsf;                                   // 16*32*64
    int*   partIdx  = (int*)(wsf + N_BATCH * NBLK * 64);     // 16*32*64
    float* boxTerm  = wsf + 2 * N_BATCH * NBLK * 64;         // 16
    float* focalCorr = boxTerm + N_BATCH;                    // 16
    float* confBase  = focalCorr + N_BATCH;                  // 16

    k_iou_argmax<<<dim3(NBLK, N_BATCH), 256, 0, stream>>>(preds, targets, partVal, partIdx);
    k_match_box<<<N_BATCH, 64, 0, stream>>>(preds, targets, partVal, partIdx,
                                            boxTerm, focalCorr, NBLK);
    k_focal_base<<<N_BATCH, 1024, 0, stream>>>(preds, confBase);
    k_finalize<<<1, 32, 0, stream>>>(boxTerm, focalCorr, confBase, out);
}